// HAN_Detector_89060441850552
// MI455X (gfx1250) — compile-verified
//
#include <hip/hip_runtime.h>
#include <hip/hip_bf16.h>
#include <math.h>

// ---------------- problem constants ----------------
#define N_TX   50000
#define N_ADDR 50000
#define NEDGE  500000
#define CCH    128     // channels
#define NH     4       // heads
#define HD     32      // dim per head
#define NG     512     // graphs
#define NOUT   2
#define NEG_SLOPE 0.2f

typedef __attribute__((ext_vector_type(2))) float v2f;
typedef __attribute__((ext_vector_type(8))) float v8f;

// ---------------- workspace layout (float offsets) ----------------
#define OFF_H_TX   ((size_t)0)
#define OFF_H_AD   ((size_t)6400000)
#define OFF_OUT_AT ((size_t)12800000)   // zero region starts here
#define OFF_OUT_TT ((size_t)19200000)
#define OFF_DEN_AT ((size_t)25600000)
#define OFF_DEN_TT ((size_t)25800000)
#define OFF_MAX_AT ((size_t)26000000)   // stored as ordered-uint keys
#define OFF_MAX_TT ((size_t)26200000)
#define OFF_PSUM   ((size_t)26400000)
#define OFF_PCNT   ((size_t)26465536)
#define OFF_SCORE  ((size_t)26466048)
#define OFF_ATTN   ((size_t)26466050)
#define ZERO_END   ((size_t)26466052)   // zero region end
#define OFF_ALS_AT ((size_t)26466052)
#define OFF_ALD_AT ((size_t)26666052)
#define OFF_ALS_TT ((size_t)26866052)
#define OFF_ALD_TT ((size_t)27066052)
#define OFF_EX_AT  ((size_t)27266052)
#define OFF_EX_TT  ((size_t)29266052)

// ---------------- helpers ----------------
__device__ __forceinline__ unsigned f2key(float f) {
    unsigned u = __float_as_uint(f);
    return (u & 0x80000000u) ? ~u : (u | 0x80000000u);
}
__device__ __forceinline__ float key2f(unsigned k) {
    return __uint_as_float((k & 0x80000000u) ? (k ^ 0x80000000u) : ~k);
}

__global__ void zero_f(float* __restrict__ p, size_t n) {
    size_t t = (size_t)blockIdx.x * blockDim.x + threadIdx.x;
    if (t < n) p[t] = 0.0f;
}

// ---------------- f32 WMMA projection: H = X[128] @ W[128x128] + b ----------------
// one wave per 16-row tile; 8 column tiles; K in steps of 4 (V_WMMA_F32_16X16X4_F32)
__global__ __launch_bounds__(32) void proj_wmma(const float* __restrict__ X,
                                                const float* __restrict__ W,
                                                const float* __restrict__ b,
                                                float* __restrict__ Hout) {
    const int lane  = threadIdx.x;
    const int m     = lane & 15;
    const int khalf = (lane >> 4) << 1;     // lanes 0-15 -> K+0/K+1, lanes 16-31 -> K+2/K+3
    const size_t row = (size_t)blockIdx.x * 16 + m;
    v8f acc[8] = {};
    const float* xrow = X + row * CCH;
    for (int k0 = 0; k0 < CCH; k0 += 4) {
        v2f a; a.x = xrow[k0 + khalf]; a.y = xrow[k0 + khalf + 1];
#pragma unroll
        for (int n = 0; n < 8; ++n) {
            const int col = n * 16 + m;
            v2f bb;
            bb.x = W[(size_t)(k0 + khalf) * CCH + col];
            bb.y = W[(size_t)(k0 + khalf + 1) * CCH + col];
            acc[n] = __builtin_amdgcn_wmma_f32_16x16x4_f32(
                false, a, false, bb, (short)0, acc[n], false, false);
        }
    }
#pragma unroll
    for (int n = 0; n < 8; ++n) {
        const int col = n * 16 + m;
        const float bias = b[col];
#pragma unroll
        for (int r = 0; r < 8; ++r) {
            const size_t orow = (size_t)blockIdx.x * 16 + ((lane < 16) ? r : r + 8);
            Hout[orow * CCH + col] = acc[n][r] + bias;
        }
    }
}

// ---------------- semantic score: sum_{n,c} q[c]*tanh((relu(OUTK) @ Wk + bk)[n,c]) ----------------
__global__ __launch_bounds__(32) void semantic_score_wmma(const float* __restrict__ OUTK,
                                                          const float* __restrict__ Wk,
                                                          const float* __restrict__ bk,
                                                          const float* __restrict__ q,
                                                          float* __restrict__ score) {
    const int lane  = threadIdx.x;
    const int m     = lane & 15;
    const int khalf = (lane >> 4) << 1;
    const size_t row = (size_t)blockIdx.x * 16 + m;
    v8f acc[8] = {};
    const float* xrow = OUTK + row * CCH;
    for (int k0 = 0; k0 < CCH; k0 += 4) {
        v2f a;
        a.x = fmaxf(xrow[k0 + khalf], 0.0f);          // relu (GAT output activation)
        a.y = fmaxf(xrow[k0 + khalf + 1], 0.0f);
#pragma unroll
        for (int n = 0; n < 8; ++n) {
            const int col = n * 16 + m;
            v2f bb;
            bb.x = Wk[(size_t)(k0 + khalf) * CCH + col];
            bb.y = Wk[(size_t)(k0 + khalf + 1) * CCH + col];
            acc[n] = __builtin_amdgcn_wmma_f32_16x16x4_f32(
                false, a, false, bb, (short)0, acc[n], false, false);
        }
    }
    float partial = 0.0f;
#pragma unroll
    for (int n = 0; n < 8; ++n) {
        const int col = n * 16 + m;
        const float qc = q[col], bc = bk[col];
#pragma unroll
        for (int r = 0; r < 8; ++r)
            partial += qc * tanhf(acc[n][r] + bc);
    }
#pragma unroll
    for (int off = 16; off > 0; off >>= 1)
        partial += __shfl_xor(partial, off, 32);
    if (lane == 0) atomicAdd(score, partial);
}

// ---------------- per-(node,head) attention logits ----------------
__global__ void alpha_logits(const float* __restrict__ Hm, const float* __restrict__ a,
                             float* __restrict__ al, int n_nodes) {
    int t = blockIdx.x * blockDim.x + threadIdx.x;
    if (t >= n_nodes * NH) return;
    const int node = t >> 2, h = t & 3;
    const float* hp = Hm + (size_t)node * CCH + h * HD;
    const float* ap = a + h * HD;
    float s = 0.0f;
#pragma unroll
    for (int d = 0; d < HD; ++d) s += hp[d] * ap[d];
    al[t] = s;
}

// ---------------- edge pass 1: segment max of leaky_relu(al_s[src]+al_d[dst]) ----------------
__global__ void edge_max(const int* __restrict__ src, const int* __restrict__ dst,
                         const float* __restrict__ als, const float* __restrict__ ald,
                         unsigned* __restrict__ amax, int ne) {
    int e = blockIdx.x * blockDim.x + threadIdx.x;
    if (e >= ne) return;
    const int s = src[e], d = dst[e];
#pragma unroll
    for (int h = 0; h < NH; ++h) {
        float a = als[s * NH + h] + ald[d * NH + h];
        a = (a > 0.0f) ? a : a * NEG_SLOPE;
        atomicMax(&amax[d * NH + h], f2key(a));
    }
}

// ---------------- edge pass 2: ex = exp(alpha - max); denom += ex ----------------
__global__ void edge_exp(const int* __restrict__ src, const int* __restrict__ dst,
                         const float* __restrict__ als, const float* __restrict__ ald,
                         const unsigned* __restrict__ amax, float* __restrict__ denom,
                         float* __restrict__ ex, int ne) {
    int e = blockIdx.x * blockDim.x + threadIdx.x;
    if (e >= ne) return;
    const int s = src[e], d = dst[e];
#pragma unroll
    for (int h = 0; h < NH; ++h) {
        float a = als[s * NH + h] + ald[d * NH + h];
        a = (a > 0.0f) ? a : a * NEG_SLOPE;
        const float ev = expf(a - key2f(amax[d * NH + h]));
        ex[(size_t)e * NH + h] = ev;
        atomicAdd(&denom[d * NH + h], ev);
    }
}

// ---------------- edge pass 3: out[dst] += h_src[src] * w  (32 lanes per edge) ----------------
__global__ void edge_scatter(const int* __restrict__ src, const int* __restrict__ dst,
                             const float* __restrict__ Hsrc, const float* __restrict__ ex,
                             const float* __restrict__ denom, float* __restrict__ out, int ne) {
    int t = blockIdx.x * blockDim.x + threadIdx.x;
    const int e = t >> 5, lane = t & 31;
    if (e >= ne) return;
    const int s = src[e], d = dst[e];
#pragma unroll
    for (int h = 0; h < NH; ++h) {
        const int c = h * HD + lane;
        const float w = ex[(size_t)e * NH + h] / (denom[d * NH + h] + 1e-16f);
        atomicAdd(&out[(size_t)d * CCH + c], Hsrc[(size_t)s * CCH + c] * w);
    }
}

// ---------------- 2-way softmax over mean scores ----------------
__global__ void attn_softmax(const float* __restrict__ score, float* __restrict__ attn) {
    const float inv_n = 1.0f / (float)N_TX;
    const float s0 = score[0] * inv_n, s1 = score[1] * inv_n;
    const float m = fmaxf(s0, s1);
    const float e0 = expf(s0 - m), e1 = expf(s1 - m);
    const float inv = 1.0f / (e0 + e1);
    attn[0] = e0 * inv;
    attn[1] = e1 * inv;
}

// ---------------- combine metapaths + mean-pool accumulation ----------------
__global__ void combine_pool(const float* __restrict__ out_at, const float* __restrict__ out_tt,
                             const float* __restrict__ attn, const int* __restrict__ batch,
                             float* __restrict__ psum, float* __restrict__ pcnt) {
    int t = blockIdx.x * blockDim.x + threadIdx.x;
    if (t >= N_TX * CCH) return;
    const int node = t >> 7, c = t & 127;
    const float f = attn[0] * fmaxf(out_at[t], 0.0f) + attn[1] * fmaxf(out_tt[t], 0.0f);
    const int g = batch[node];
    atomicAdd(&psum[(size_t)g * CCH + c], f);
    if (c == 0) atomicAdd(&pcnt[g], 1.0f);
}

// ---------------- final linear: [G,128] @ [128,2] + bl ----------------
__global__ void final_linear(const float* __restrict__ psum, const float* __restrict__ pcnt,
                             const float* __restrict__ Wl, const float* __restrict__ bl,
                             float* __restrict__ outp) {
    int t = blockIdx.x * blockDim.x + threadIdx.x;
    if (t >= NG * NOUT) return;
    const int g = t >> 1, o = t & 1;
    const float inv_cnt = 1.0f / fmaxf(pcnt[g], 1.0f);
    float s = 0.0f;
#pragma unroll
    for (int c = 0; c < CCH; ++c)
        s += psum[(size_t)g * CCH + c] * Wl[c * NOUT + o];
    outp[t] = s * inv_cnt + bl[o];
}

// ---------------- launch ----------------
extern "C" void kernel_launch(void* const* d_in, const int* in_sizes, int n_in,
                              void* d_out, int out_size, void* d_ws, size_t ws_size,
                              hipStream_t stream) {
    (void)in_sizes; (void)n_in; (void)out_size; (void)ws_size;

    const float* x_tx   = (const float*)d_in[0];
    const float* x_addr = (const float*)d_in[1];
    // d_in[2] = ei_ta : out_ta is dead code in the reference -> skipped
    const int*   ei_at  = (const int*)d_in[3];
    const int*   ei_tt  = (const int*)d_in[4];
    const int*   batch  = (const int*)d_in[5];
    const float* W_tx   = (const float*)d_in[6];
    const float* b_tx   = (const float*)d_in[7];
    const float* W_ad   = (const float*)d_in[8];
    const float* b_ad   = (const float*)d_in[9];
    const float* a_src_at = (const float*)d_in[12];
    const float* a_dst_at = (const float*)d_in[13];
    const float* a_src_tt = (const float*)d_in[14];
    const float* a_dst_tt = (const float*)d_in[15];
    const float* Wk = (const float*)d_in[16];
    const float* bk = (const float*)d_in[17];
    const float* q  = (const float*)d_in[18];
    const float* Wl = (const float*)d_in[19];
    const float* bl = (const float*)d_in[20];

    float* ws = (float*)d_ws;
    float*    h_tx   = ws + OFF_H_TX;
    float*    h_ad   = ws + OFF_H_AD;
    float*    out_at = ws + OFF_OUT_AT;
    float*    out_tt = ws + OFF_OUT_TT;
    float*    den_at = ws + OFF_DEN_AT;
    float*    den_tt = ws + OFF_DEN_TT;
    unsigned* max_at = (unsigned*)(ws + OFF_MAX_AT);
    unsigned* max_tt = (unsigned*)(ws + OFF_MAX_TT);
    float*    psum   = ws + OFF_PSUM;
    float*    pcnt   = ws + OFF_PCNT;
    float*    score  = ws + OFF_SCORE;
    float*    attn   = ws + OFF_ATTN;
    float*    als_at = ws + OFF_ALS_AT;
    float*    ald_at = ws + OFF_ALD_AT;
    float*    als_tt = ws + OFF_ALS_TT;
    float*    ald_tt = ws + OFF_ALD_TT;
    float*    ex_at  = ws + OFF_EX_AT;
    float*    ex_tt  = ws + OFF_EX_TT;

    // 1. zero accumulators (out/denom/max-keys/pool/score)
    {
        size_t nz = ZERO_END - OFF_OUT_AT;
        zero_f<<<(unsigned)((nz + 255) / 256), 256, 0, stream>>>(out_at, nz);
    }

    // 2. per-type projections (f32 WMMA)
    proj_wmma<<<N_TX / 16, 32, 0, stream>>>(x_tx, W_tx, b_tx, h_tx);
    proj_wmma<<<N_ADDR / 16, 32, 0, stream>>>(x_addr, W_ad, b_ad, h_ad);

    // 3. attention logits
    alpha_logits<<<(N_ADDR * NH + 255) / 256, 256, 0, stream>>>(h_ad, a_src_at, als_at, N_ADDR);
    alpha_logits<<<(N_TX * NH + 255) / 256, 256, 0, stream>>>(h_tx, a_dst_at, ald_at, N_TX);
    alpha_logits<<<(N_TX * NH + 255) / 256, 256, 0, stream>>>(h_tx, a_src_tt, als_tt, N_TX);
    alpha_logits<<<(N_TX * NH + 255) / 256, 256, 0, stream>>>(h_tx, a_dst_tt, ald_tt, N_TX);

    // 4-6. edge softmax + scatter, addr->tx and tx->tx
    const int egrid = (NEDGE + 255) / 256;
    const int sgrid = (NEDGE * 32) / 256;
    edge_max<<<egrid, 256, 0, stream>>>(ei_at, ei_at + NEDGE, als_at, ald_at, max_at, NEDGE);
    edge_max<<<egrid, 256, 0, stream>>>(ei_tt, ei_tt + NEDGE, als_tt, ald_tt, max_tt, NEDGE);
    edge_exp<<<egrid, 256, 0, stream>>>(ei_at, ei_at + NEDGE, als_at, ald_at, max_at, den_at, ex_at, NEDGE);
    edge_exp<<<egrid, 256, 0, stream>>>(ei_tt, ei_tt + NEDGE, als_tt, ald_tt, max_tt, den_tt, ex_tt, NEDGE);
    edge_scatter<<<sgrid, 256, 0, stream>>>(ei_at, ei_at + NEDGE, h_ad, ex_at, den_at, out_at, NEDGE);
    edge_scatter<<<sgrid, 256, 0, stream>>>(ei_tt, ei_tt + NEDGE, h_tx, ex_tt, den_tt, out_tt, NEDGE);

    // 7. semantic attention scores (f32 WMMA fused with tanh/q-dot)
    semantic_score_wmma<<<N_TX / 16, 32, 0, stream>>>(out_at, Wk, bk, q, score + 0);
    semantic_score_wmma<<<N_TX / 16, 32, 0, stream>>>(out_tt, Wk, bk, q, score + 1);
    attn_softmax<<<1, 1, 0, stream>>>(score, attn);

    // 8. combine + mean pool + final linear
    combine_pool<<<(N_TX * CCH + 255) / 256, 256, 0, stream>>>(out_at, out_tt, attn, batch, psum, pcnt);
    final_linear<<<(NG * NOUT + 255) / 256, 256, 0, stream>>>(psum, pcnt, Wl, bl, (float*)d_out);
}